// LocalGrouper_12627203850343
// MI455X (gfx1250) — compile-verified
//
// LocalGrouper for MI455X (gfx1250, wave32).
// Pipeline: key init -> exact global quantile select (16+16 bit histograms) ->
// 4x stable LSD radix sort (hist/scan/scatter, ballot ranking) -> tiny
// planner (bin counts via binary search + renorm loop) -> row gather using
// CDNA5 async global<->LDS DMA (ASYNCcnt) when available.
#include <hip/hip_runtime.h>
#include <stdint.h>

#define BATCH  8
#define NPTS   65536
#define CDIM   64
#define NBINS  6
#define MOUT   16384           // NPTS/4
#define TOTALN (BATCH * NPTS)  // 524288
#define TILE   1024
#define BLKB   (NPTS / TILE)   // 64 blocks per batch
#define HISTN  (BATCH * 256 * BLKB)

// ---------------- key init: order-preserving uint transform -----------------
// kt ascending == score descending; radix sort ascending on kt is stable.
__global__ void init_keys(const float* __restrict__ score,
                          unsigned* __restrict__ keys,
                          unsigned* __restrict__ idx) {
  unsigned i = blockIdx.x * 256u + threadIdx.x;
  unsigned u = __float_as_uint(score[i]);
  u ^= (u & 0x80000000u) ? 0xFFFFFFFFu : 0x80000000u;  // ascending-order bits
  keys[i] = ~u;                                        // descending score
  idx[i]  = i & (NPTS - 1);                            // index within batch
}

// ---------------- radix pass 1/3: per-block digit histogram ----------------
__global__ void radix_hist(const unsigned* __restrict__ keys,
                           unsigned* __restrict__ h, int shift) {
  __shared__ unsigned lh[256];
  int tid = threadIdx.x;
  lh[tid] = 0;
  __syncthreads();
  int b = blockIdx.x >> 6, t = blockIdx.x & 63;
  size_t base = (size_t)b * NPTS + (size_t)t * TILE;
  for (int c = 0; c < 4; ++c) {
    unsigned k = keys[base + c * 256 + tid];
    atomicAdd(&lh[(k >> shift) & 255u], 1u);           // ds_add_u32
  }
  __syncthreads();
  h[((size_t)b * 256 + tid) * BLKB + t] = lh[tid];
}

// ---------------- radix pass 2/3: per-batch digit-major exclusive scan -----
__global__ void radix_scan(unsigned* __restrict__ h) {
  __shared__ unsigned sums[1024];
  int b = blockIdx.x, tid = threadIdx.x;
  unsigned base = (unsigned)b * (256u * BLKB) + tid * 16u;
  unsigned v[16], s = 0;
  for (int q = 0; q < 16; ++q) v[q] = h[base + q];
  for (int q = 0; q < 16; ++q) { unsigned x = v[q]; v[q] = s; s += x; }
  sums[tid] = s;
  __syncthreads();
  for (int off = 1; off < 1024; off <<= 1) {
    unsigned x = (tid >= off) ? sums[tid - off] : 0u;
    __syncthreads();
    sums[tid] += x;
    __syncthreads();
  }
  unsigned ex = (tid == 0) ? 0u : sums[tid - 1];
  for (int q = 0; q < 16; ++q) h[base + q] = v[q] + ex;
}

// ---------------- radix pass 3/3: stable scatter (wave32 ballot ranking) ---
__global__ void radix_scatter(const unsigned* __restrict__ skeys,
                              const unsigned* __restrict__ sidx,
                              unsigned* __restrict__ dkeys,
                              unsigned* __restrict__ didx,
                              const unsigned* __restrict__ h, int shift) {
  __shared__ unsigned digitBase[256];
  __shared__ unsigned runBase[256];
  __shared__ unsigned chunkTot[256];
  __shared__ unsigned waveCnt[8 * 256];
  int tid = threadIdx.x;
  int b = blockIdx.x >> 6, t = blockIdx.x & 63;
  digitBase[tid] = h[((size_t)b * 256 + tid) * BLKB + t];
  runBase[tid] = 0;
  int lane = tid & 31, wave = tid >> 5;  // wave32
  size_t base = (size_t)b * NPTS + (size_t)t * TILE;
  for (int c = 0; c < 4; ++c) {
    unsigned k = skeys[base + c * 256 + tid];
    unsigned v = sidx[base + c * 256 + tid];
    unsigned d = (k >> shift) & 255u;
    unsigned match = 0xFFFFFFFFu;
#pragma unroll
    for (int bit = 0; bit < 8; ++bit) {
      unsigned bb = (unsigned)__ballot((d >> bit) & 1u);
      match &= ((d >> bit) & 1u) ? bb : ~bb;
    }
    unsigned rankW = __popc(match & ((1u << lane) - 1u));
    unsigned cntW  = __popc(match);
    __syncthreads();  // protect prior-iteration LDS reads
#pragma unroll
    for (int w = 0; w < 8; ++w) waveCnt[w * 256 + tid] = 0u;
    __syncthreads();
    waveCnt[wave * 256 + d] = cntW;  // benign same-value race within a wave
    __syncthreads();
    unsigned s = 0;
#pragma unroll
    for (int w = 0; w < 8; ++w) {
      unsigned x = waveCnt[w * 256 + tid];
      waveCnt[w * 256 + tid] = s;
      s += x;
    }
    chunkTot[tid] = s;
    __syncthreads();
    unsigned pos = digitBase[d] + runBase[d] + waveCnt[wave * 256 + d] + rankW;
    dkeys[(size_t)b * NPTS + pos] = k;
    didx[(size_t)b * NPTS + pos]  = v;
    __syncthreads();
    runBase[tid] += chunkTot[tid];
  }
}

// ---------------- exact global quantile select: high 16 bits ---------------
__global__ void hist_hi(const unsigned* __restrict__ keys,
                        unsigned* __restrict__ gh) {
  unsigned i = blockIdx.x * 256u + threadIdx.x;
  atomicAdd(&gh[keys[i] >> 16], 1u);
}

__global__ void select_hi(const unsigned* __restrict__ gh,
                          unsigned* __restrict__ bsel) {
  __shared__ unsigned sums[1024];
  int tid = threadIdx.x;
  unsigned base = tid * 64u, s = 0;
  for (int q = 0; q < 64; ++q) s += gh[base + q];
  sums[tid] = s;
  __syncthreads();
  for (int off = 1; off < 1024; off <<= 1) {
    unsigned x = (tid >= off) ? sums[tid - off] : 0u;
    __syncthreads();
    sums[tid] += x;
    __syncthreads();
  }
  unsigned ex = (tid == 0) ? 0u : sums[tid - 1];
  unsigned inc = sums[tid];
  for (int i = 0; i < 5; ++i) {
    unsigned r = (unsigned)((float)(i + 1) / 6.0f * 524288.0f);  // matches jnp f32
    if (r >= ex && r < inc) {
      unsigned c = ex;
      for (int q = 0; q < 64; ++q) {
        unsigned cnt = gh[base + q];
        if (r < c + cnt) { bsel[i] = base + q; bsel[8 + i] = r - c; break; }
        c += cnt;
      }
    }
  }
}

// ---------------- exact global quantile select: low 16 bits ----------------
__global__ void hist_lo(const unsigned* __restrict__ keys,
                        const unsigned* __restrict__ bsel,
                        unsigned* __restrict__ gl) {
  __shared__ unsigned sb[5];
  if (threadIdx.x < 5) sb[threadIdx.x] = bsel[threadIdx.x];
  __syncthreads();
  unsigned i = blockIdx.x * 256u + threadIdx.x;
  unsigned k = keys[i];
  unsigned hi = k >> 16, lo = k & 0xFFFFu;
#pragma unroll
  for (int q = 0; q < 5; ++q)
    if (hi == sb[q]) atomicAdd(&gl[q * 65536 + lo], 1u);
}

__global__ void select_lo(const unsigned* __restrict__ gl,
                          const unsigned* __restrict__ bsel,
                          unsigned* __restrict__ bnd) {
  __shared__ unsigned sums[1024];
  int tid = threadIdx.x;
  for (int i = 0; i < 5; ++i) {
    const unsigned* g = gl + (size_t)i * 65536;
    unsigned base = tid * 64u, s = 0;
    for (int q = 0; q < 64; ++q) s += g[base + q];
    sums[tid] = s;
    __syncthreads();
    for (int off = 1; off < 1024; off <<= 1) {
      unsigned x = (tid >= off) ? sums[tid - off] : 0u;
      __syncthreads();
      sums[tid] += x;
      __syncthreads();
    }
    unsigned ex = (tid == 0) ? 0u : sums[tid - 1];
    unsigned inc = sums[tid];
    unsigned r = bsel[8 + i];
    if (r >= ex && r < inc) {
      unsigned c = ex;
      for (int q = 0; q < 64; ++q) {
        unsigned cnt = g[base + q];
        if (r < c + cnt) { bnd[i] = (bsel[i] << 16) | (base + q); break; }
        c += cnt;
      }
    }
    __syncthreads();  // sums reused next i
  }
}

// ---------------- planner: bin counts + renormalization -> segments --------
__global__ void planner(const unsigned* __restrict__ keys,  // sorted kt per batch
                        const float* __restrict__ bin_prob,
                        const unsigned* __restrict__ bnd,   // [5] kt boundaries
                        unsigned* __restrict__ segOff,      // [BATCH*8]
                        unsigned* __restrict__ outStart) {  // [BATCH*8]
  __shared__ unsigned s_bnd[5];
  __shared__ unsigned s_cum[BATCH][8];
  int tid = threadIdx.x;
  if (tid < 5) s_bnd[tid] = bnd[tid];
  __syncthreads();
  if (tid < BATCH * NBINS) {  // 48 parallel binary searches (upper_bound)
    int b = tid / NBINS, j = tid % NBINS;
    unsigned c;
    if (j < 5) {
      const unsigned* a = keys + (size_t)b * NPTS;
      unsigned v = s_bnd[j], lo = 0, hi = NPTS;
      while (lo < hi) {
        unsigned mid = (lo + hi) >> 1;
        if (a[mid] <= v) lo = mid + 1; else hi = mid;
      }
      c = lo;
    } else {
      c = NPTS;
    }
    s_cum[b][j + 1] = c;
    if (j == 0) s_cum[b][0] = 0;
  }
  __syncthreads();
  if (tid < BATCH) {  // replicate reference f32 renorm loop exactly
    int b = tid;
    float mn[NBINS], p[NBINS], ch[NBINS];
    for (int q = 0; q < NBINS; ++q) {
      mn[q] = (float)(s_cum[b][q + 1] - s_cum[b][q]);
      p[q]  = bin_prob[b * NBINS + q] * mn[q] + 1e-10f;
      ch[q] = 0.0f;
    }
    for (int it = 0; it < NBINS; ++it) {
      float s = 0.0f;
      for (int q = 0; q < NBINS; ++q) s += p[q];
      float div = (s == 0.0f) ? 1.0f : s;
      for (int q = 0; q < NBINS; ++q) p[q] = p[q] / div;
      float cs = 0.0f;
      for (int q = 0; q < NBINS; ++q) cs += ch[q];
      float need = 16384.0f - cs;
      for (int q = 0; q < NBINS; ++q) {
        ch[q] += p[q] * need;
        if (ch[q] > mn[q]) ch[q] = mn[q];
      }
      for (int q = 0; q < NBINS; ++q) p[q] = (ch[q] < mn[q]) ? p[q] : 0.0f;
    }
    int k[NBINS], sum = 0;
    for (int q = 0; q < NBINS; ++q) { k[q] = (int)ch[q]; sum += k[q]; }
    int deficit = MOUT - sum;
    float best = -3.4e38f; int jb = 0;
    for (int q = 0; q < NBINS; ++q) {
      float v = mn[q] - (float)k[q];
      if (v > best) { best = v; jb = q; }  // first max, like jnp.argmax
    }
    k[jb] += deficit;
    int acc = 0;
    for (int q = 0; q < NBINS; ++q) {
      outStart[b * 8 + q] = (unsigned)acc;
      segOff[b * 8 + q]   = s_cum[b][q];
      acc += k[q];
    }
    outStart[b * 8 + 6] = (unsigned)acc;   // == MOUT
    segOff[b * 8 + 6]   = s_cum[b][6];
  }
}

// ---------------- final gather: 16B per thread, async DMA via LDS ----------
#if defined(__HIP_DEVICE_COMPILE__) &&                                        \
    __has_builtin(__builtin_amdgcn_global_load_async_to_lds_b128) &&          \
    __has_builtin(__builtin_amdgcn_global_store_async_from_lds_b128)
#define USE_ASYNC_LDS 1
typedef int v4i __attribute__((vector_size(16)));
typedef v4i __attribute__((address_space(1))) as1_v4i;  // global
typedef v4i __attribute__((address_space(3))) as3_v4i;  // LDS
#else
#define USE_ASYNC_LDS 0
#endif

__global__ void gather_rows(const float* __restrict__ points,
                            const unsigned* __restrict__ sortedIdx,
                            const unsigned* __restrict__ segOff,
                            const unsigned* __restrict__ outStart,
                            float* __restrict__ out) {
  unsigned g = blockIdx.x * 256u + threadIdx.x;  // over BATCH*MOUT*16 float4s
  unsigned b = g >> 18;                          // MOUT*16 = 2^18
  unsigned rem = g & 0x3FFFFu;
  unsigned m = rem >> 4;
  unsigned f4 = rem & 15u;
  const unsigned* os = outStart + b * 8;
  unsigned j = 0;
#pragma unroll
  for (int q = 1; q <= 5; ++q) j += (m >= os[q]) ? 1u : 0u;  // searchsorted
  unsigned pos = segOff[b * 8 + j] + (m - os[j]);
  unsigned src = sortedIdx[(size_t)b * NPTS + pos];
  const float* sp = points + ((size_t)b * NPTS + src) * CDIM + f4 * 4u;
  float* dp = out + ((size_t)b * MOUT + m) * CDIM + f4 * 4u;
#if USE_ASYNC_LDS
  // CDNA5 async DMA: global -> LDS -> global, data never touches VGPRs.
  __shared__ __align__(16) unsigned char stage[256 * 16];
  as3_v4i* lp = (as3_v4i*)(stage + threadIdx.x * 16);
  __builtin_amdgcn_global_load_async_to_lds_b128((as1_v4i*)sp, lp, 0, 0);
#if __has_builtin(__builtin_amdgcn_s_wait_asynccnt)
  __builtin_amdgcn_s_wait_asynccnt(0);
#else
  asm volatile("s_wait_asynccnt 0x0" ::: "memory");
#endif
  __builtin_amdgcn_global_store_async_from_lds_b128((as1_v4i*)dp, lp, 0, 0);
  // s_endpgm performs implicit wait-idle; store completion is guaranteed.
#else
  float4 v = *(const float4*)sp;
  *(float4*)dp = v;
#endif
}

// ---------------- driver ----------------------------------------------------
extern "C" void kernel_launch(void* const* d_in, const int* in_sizes, int n_in,
                              void* d_out, int out_size, void* d_ws,
                              size_t ws_size, hipStream_t stream) {
  (void)in_sizes; (void)n_in; (void)out_size; (void)ws_size;
  const float* score   = (const float*)d_in[0];  // (8,1,65536)
  const float* points  = (const float*)d_in[1];  // (8,65536,64)
  const float* binprob = (const float*)d_in[2];  // (8,6)
  float* out = (float*)d_out;                    // (8,16384,64)

  unsigned* ws    = (unsigned*)d_ws;             // ~10.5 MB used
  unsigned* keysA = ws;
  unsigned* idxA  = keysA + TOTALN;
  unsigned* keysB = idxA + TOTALN;
  unsigned* idxB  = keysB + TOTALN;
  unsigned* bhist = idxB + TOTALN;               // HISTN
  unsigned* gh    = bhist + HISTN;               // 65536
  unsigned* gl    = gh + 65536;                  // 5*65536
  unsigned* bsel  = gl + 5 * 65536;              // 16
  unsigned* bnd   = bsel + 16;                   // 8
  unsigned* segOf = bnd + 8;                     // BATCH*8
  unsigned* outSt = segOf + BATCH * 8;           // BATCH*8

  init_keys<<<TOTALN / 256, 256, 0, stream>>>(score, keysA, idxA);

  // Exact global quantile boundaries on the (unsorted) key multiset.
  (void)hipMemsetAsync(gh, 0, 65536 * sizeof(unsigned), stream);
  (void)hipMemsetAsync(gl, 0, 5 * 65536 * sizeof(unsigned), stream);
  hist_hi<<<TOTALN / 256, 256, 0, stream>>>(keysA, gh);
  select_hi<<<1, 1024, 0, stream>>>(gh, bsel);
  hist_lo<<<TOTALN / 256, 256, 0, stream>>>(keysA, bsel, gl);
  select_lo<<<1, 1024, 0, stream>>>(gl, bsel, bnd);

  // 4-pass stable LSD radix sort per batch (key, index).
  unsigned *sk = keysA, *si = idxA, *dk = keysB, *di = idxB;
  for (int p = 0; p < 4; ++p) {
    radix_hist<<<BATCH * BLKB, 256, 0, stream>>>(sk, bhist, p * 8);
    radix_scan<<<BATCH, 1024, 0, stream>>>(bhist);
    radix_scatter<<<BATCH * BLKB, 256, 0, stream>>>(sk, si, dk, di, bhist, p * 8);
    unsigned* tmp;
    tmp = sk; sk = dk; dk = tmp;
    tmp = si; si = di; di = tmp;
  }
  // After an even number of passes, sorted data is back in keysA/idxA.

  planner<<<1, 256, 0, stream>>>(sk, binprob, bnd, segOf, outSt);
  gather_rows<<<(BATCH * MOUT * 16) / 256, 256, 0, stream>>>(points, si, segOf,
                                                             outSt, out);
}